// XEyTransformerLayer_85512798863886
// MI455X (gfx1250) — compile-verified
//
#include <hip/hip_runtime.h>
#include <math.h>

typedef float v2f __attribute__((ext_vector_type(2)));
typedef float v8f __attribute__((ext_vector_type(8)));

constexpr int NN = 256;           // nodes
constexpr int EM_WAVES = 4;
constexpr int EM_THREADS = 128;   // 4 wave32

// ---------------- workspace offsets (floats) ----------------
constexpr size_t OFF_XE1 = 0;              // 4*256*64
constexpr size_t OFF_XE2 = 65536;
constexpr size_t OFF_FE1 = 131072;
constexpr size_t OFF_FE2 = 196608;
constexpr size_t OFF_QM  = 262144;         // 4*256*256
constexpr size_t OFF_KM  = 524288;
constexpr size_t OFF_VV  = 786432;
constexpr size_t OFF_QF  = 1048576;
constexpr size_t OFF_FXM = 1310720;
constexpr size_t OFF_FXA = 1572864;
constexpr size_t OFF_POSM = 1835008;       // 4*256*3
constexpr size_t OFF_PMM  = 1838080;
constexpr size_t OFF_NPOS = 1841152;
constexpr size_t OFF_NPP  = 1844224;
constexpr size_t OFF_NORMPOS = 1847296;    // 4*256
constexpr size_t OFF_NORMP   = 1848320;
constexpr size_t OFF_YEA = 1849344;        // 4*64
constexpr size_t OFF_YEM = 1849600;
constexpr size_t OFF_YXM = 1849856;        // 4*256
constexpr size_t OFF_YXA = 1850880;
constexpr size_t OFF_ABUF = 1851904;       // 4*256*256*8
constexpr size_t OFF_EXST = 3949056;       // 4*256*3*5*64
constexpr size_t OFF_EYST = 4932096;       // 4*256*2*5*64
constexpr size_t OFF_VEL  = 5587456;       // 4*256*3
constexpr size_t OFF_VELP = 5590528;
constexpr size_t OFF_WV   = 5593600;       // 4*256*2048
constexpr size_t OFF_NEWX = 7690752;       // 4*256*256
constexpr size_t OFF_QP = 7952896;         // 4*256*64
constexpr size_t OFF_KP = 8018432;
constexpr size_t OFF_VP = 8083968;
constexpr size_t OFF_WVPO = 8149504;       // 4*256*3
constexpr size_t OFF_WVO  = 8152576;       // 4*256*256 (out(wV), ungated)

// ---------------- output offsets (floats) ----------------
constexpr size_t OOUT_X   = 0;             // 262144
constexpr size_t OOUT_E   = 262144;        // 16777216
constexpr size_t OOUT_Y   = 17039360;      // 512
constexpr size_t OOUT_F   = 17039872;      // 262144
constexpr size_t OOUT_VEL = 17302016;      // 3072
constexpr size_t OOUT_VELP= 17305088;      // 3072

struct Ctx {
  // inputs
  const float *X,*E,*y,*pos,*f,*p,*nmask,*pmask;
  // params (flattened in _make_params insertion order)
  const float *inE_w,*inE_b;
  const float *xem1_w,*xem1_b,*xem2_w,*xem2_b;
  const float *fem1_w,*fem1_b,*fem2_w,*fem2_b;
  const float *ld1_w,*ld1_b,*lnp1_w,*lnp1_b,*lnp2_w,*lnp2_b;
  const float *dae_w,*dae_b,*dme_w,*dme_b;
  const float *ld1p_w,*ld1p_b,*lnp_w,*lnp_b,*lnp2p_w,*lnp2p_b;
  const float *daep_w,*daep_b,*dmep_w,*dmep_b;
  const float *k_w,*k_b,*q_w,*q_b,*v_w,*v_b;
  const float *a_w;
  const float *out_w,*out_b;
  const float *qf_w,*qf_b;
  const float *af_w;
  const float *kpos_w,*kpos_b,*qpos_w,*qpos_b,*vpos_w,*vpos_b;
  const float *apos_w;
  const float *dpos_w,*dpos_b;
  const float *outpos_w,*outpos_b;
  const float *eatt_w,*eatt_b,*posatt_w,*posatt_b,*patt_w,*patt_b;
  const float *exm_w,*exm_b,*pxm_w,*pxm_b,*ppxm_w,*ppxm_b;
  const float *yem_w,*yem_b,*yea_w,*yea_b;
  const float *yxm_w,*yxm_b,*yxa_w,*yxa_b;
  const float *fxm_w,*fxm_b,*fxa_w,*fxa_b;
  const float *yy_w,*yy_b,*xy_w,*xy_b,*ey_w,*ey_b,*dy_w,*dy_b;
  const float *ep1_w,*ep2_w,*epp1_w,*epp2_w;
  const float *xout_w,*xout_b;
  const float *eout_w,*eout_b;
  const float *y1_w,*y1_b,*y2_w,*y2_b;
  // workspace
  float *xe1,*xe2,*fe1,*fe2,*Qm,*Km,*Vv,*Qf,*fxmv,*fxav;
  float *posm,*pmm,*npos,*npp,*normpos,*normp;
  float *yeav,*yemv,*yxmv,*yxav;
  float *abuf,*exst,*eyst,*vel,*velp,*wv,*newx,*Qp,*Kp,*Vp,*wvpo,*wvo;
  // outputs
  float *oX,*oE,*oy,*ovel,*ovelp;
};

// ---------------- WMMA helpers (V_WMMA_F32_16X16X4_F32) ----------------
static __device__ __forceinline__ v8f wmma4(v2f a, v2f b, v8f c) {
  return __builtin_amdgcn_wmma_f32_16x16x4_f32(false, a, false, b, (short)0, c,
                                               false, false);
}

// out(16x64) += in(16x64, row-major, stride 64) @ W(64x64 row-major)
// A frag: lane L holds (m=L&15, k=kb+2*(L>>4)+{0,1})
// B frag: lane L holds (k=kb+2*(L>>4)+{0,1}, n=nt*16+(L&15))
// C/D:    vgpr r holds (m=r+8*(L>>4), n=nt*16+(L&15))
static __device__ __forceinline__ void gemm_tile(const float* __restrict__ in,
                                                 const float* __restrict__ W,
                                                 v8f acc[4], int lane) {
  const int mA = lane & 15;
  const int kh2 = (lane >> 4) << 1;
  const int nl = lane & 15;
  for (int kb = 0; kb < 64; kb += 4) {
    v2f a;
    a.x = in[mA * 64 + kb + kh2];
    a.y = in[mA * 64 + kb + kh2 + 1];
#pragma unroll
    for (int nt = 0; nt < 4; ++nt) {
      v2f bf;
      bf.x = W[(kb + kh2) * 64 + nt * 16 + nl];
      bf.y = W[(kb + kh2 + 1) * 64 + nt * 16 + nl];
      acc[nt] = wmma4(a, bf, acc[nt]);
    }
  }
}

static __device__ __forceinline__ void init_acc(v8f acc[4], const float* bias,
                                                int nl) {
#pragma unroll
  for (int nt = 0; nt < 4; ++nt) {
    float bb = bias ? bias[nt * 16 + nl] : 0.f;
    v8f t;
#pragma unroll
    for (int r = 0; r < 8; ++r) t[r] = bb;
    acc[nt] = t;
  }
}

// masked statistics: sum, masked sum, masked sumsq, shifted min/max
struct St { float s, sm, sq, mn, mx; };
static __device__ __forceinline__ void st_init(St& a) {
  a.s = 0.f; a.sm = 0.f; a.sq = 0.f; a.mn = 1e30f; a.mx = -1e30f;
}
static __device__ __forceinline__ void st_upd(St& a, float v, float mm) {
  a.s += v; a.sm += v * mm; a.sq += v * v * mm;
  a.mn = fminf(a.mn, v + 1.0e6f * (1.f - mm));
  a.mx = fmaxf(a.mx, v - 1.0e6f * (1.f - mm));
}

// ================= per-batch y-derived vectors =================
__global__ __launch_bounds__(256) void batch_pre_kernel(Ctx c) {
  __shared__ float sy[128];
  int b = blockIdx.x, t = threadIdx.x;
  if (t < 128) sy[t] = c.y[b * 128 + t];
  __syncthreads();
  if (t < 64) {
    float am = c.yem_b[t], aa = c.yea_b[t];
    for (int d = 0; d < 128; ++d) {
      am += sy[d] * c.yem_w[d * 64 + t];
      aa += sy[d] * c.yea_w[d * 64 + t];
    }
    c.yemv[b * 64 + t] = am;
    c.yeav[b * 64 + t] = aa;
  }
  float xm = c.yxm_b[t], xa = c.yxa_b[t];
  for (int d = 0; d < 128; ++d) {
    xm += sy[d] * c.yxm_w[d * 256 + t];
    xa += sy[d] * c.yxa_w[d * 256 + t];
  }
  c.yxmv[b * 256 + t] = xm;
  c.yxav[b * 256 + t] = xa;
}

// ================= per-node precompute =================
__global__ __launch_bounds__(256) void node_pre_kernel(Ctx c) {
  __shared__ float sX[256], sF[256];
  int bi = blockIdx.x, t = threadIdx.x;
  sX[t] = c.X[(size_t)bi * 256 + t];
  sF[t] = c.f[(size_t)bi * 256 + t];
  float m = c.nmask[bi], pm = c.pmask[bi];
  if (t == 0) {
    float pr[3], pp[3], s = 0.f, sp = 0.f;
    for (int q3 = 0; q3 < 3; ++q3) {
      pr[q3] = c.pos[bi * 3 + q3] * m;
      pp[q3] = c.p[bi * 3 + q3] * pm;
      s += pr[q3] * pr[q3];
      sp += pp[q3] * pp[q3];
    }
    float nr = sqrtf(fmaxf(s, 1e-12f)), nq = sqrtf(fmaxf(sp, 1e-12f));
    c.normpos[bi] = nr; c.normp[bi] = nq;
    for (int q3 = 0; q3 < 3; ++q3) {
      c.posm[bi * 3 + q3] = pr[q3];
      c.npos[bi * 3 + q3] = pr[q3] / (nr + 1e-7f);
      c.pmm[bi * 3 + q3] = pp[q3];
      c.npp[bi * 3 + q3] = pp[q3] / (nq + 1e-7f);
    }
  }
  __syncthreads();
  {  // de-wide gating vectors
    int grp = t >> 6, ch = t & 63;
    const float *w8, *b8, *in8; float mm; float* dst;
    if (grp == 0)      { w8 = c.xem1_w; b8 = c.xem1_b; in8 = sX; mm = m;  dst = c.xe1; }
    else if (grp == 1) { w8 = c.xem2_w; b8 = c.xem2_b; in8 = sX; mm = m;  dst = c.xe2; }
    else if (grp == 2) { w8 = c.fem1_w; b8 = c.fem1_b; in8 = sF; mm = pm; dst = c.fe1; }
    else               { w8 = c.fem2_w; b8 = c.fem2_b; in8 = sF; mm = pm; dst = c.fe2; }
    float a = b8[ch];
    for (int d = 0; d < 256; ++d) a += in8[d] * w8[d * 64 + ch];
    dst[(size_t)bi * 64 + ch] = a * mm;
  }
  // dx-wide projections
  float aq = c.q_b[t], ak = c.k_b[t], av = c.v_b[t];
  float aqf = c.qf_b[t], afm = c.fxm_b[t], afa = c.fxa_b[t];
  for (int d = 0; d < 256; ++d) {
    float xv = sX[d], fv = sF[d];
    aq  += xv * c.q_w[d * 256 + t];
    ak  += xv * c.k_w[d * 256 + t];
    av  += xv * c.v_w[d * 256 + t];
    aqf += fv * c.qf_w[d * 256 + t];
    afm += fv * c.fxm_w[d * 256 + t];
    afa += fv * c.fxa_w[d * 256 + t];
  }
  c.Qm[(size_t)bi * 256 + t] = aq * m;
  c.Km[(size_t)bi * 256 + t] = ak * m;
  c.Vv[(size_t)bi * 256 + t] = av * m;
  c.Qf[(size_t)bi * 256 + t] = aqf * pm;
  c.fxmv[(size_t)bi * 256 + t] = afm;
  c.fxav[(size_t)bi * 256 + t] = afa;
}

// ================= fused E pipeline (WMMA) =================
__global__ __launch_bounds__(EM_THREADS) void e_main_kernel(Ctx c) {
  __shared__ float sD1[EM_WAVES][1024];
  __shared__ float sD1P[EM_WAVES][1024];
  __shared__ float sYC[EM_WAVES][1024];
  __shared__ float sEN[EM_WAVES][1024];
  __shared__ float sQmw[2048];
  __shared__ float sQfw[2048];
  __shared__ float sJ[EM_WAVES][16][12];
  __shared__ float sStats[EM_WAVES][5][5][64];
  __shared__ float sVel[3], sVelp[3];

  const int tid = threadIdx.x;
  const int wave = tid >> 5;
  const int lane = tid & 31;
  const int bi = blockIdx.x;
  const int b = bi >> 8;

  if (tid < 3) { sVel[tid] = 0.f; sVelp[tid] = 0.f; }
  for (int e = tid; e < 2048; e += EM_THREADS) {
    int d = e >> 3, h = e & 7;
    sQmw[e] = c.Qm[(size_t)bi * 256 + d] * c.a_w[d * 8 + h];
    sQfw[e] = c.Qf[(size_t)bi * 256 + d] * c.af_w[d * 8 + h];
  }

  const float m_i = c.nmask[bi];
  const float pm_i = c.pmask[bi];
  float posi[3], npi[3], ppi[3], nppi[3];
#pragma unroll
  for (int q3 = 0; q3 < 3; ++q3) {
    posi[q3] = c.posm[bi * 3 + q3];
    npi[q3]  = c.npos[bi * 3 + q3];
    ppi[q3]  = c.pmm[bi * 3 + q3];
    nppi[q3] = c.npp[bi * 3 + q3];
  }
  const float normpos_i = c.normpos[bi];
  const float normp_i = c.normp[bi];

  const int nl = lane & 15;
  const int khh = lane >> 4;
  float xe2c[4], fe2c[4], yeac[4], yemc[4];
#pragma unroll
  for (int nt = 0; nt < 4; ++nt) {
    int ch = nt * 16 + nl;
    xe2c[nt] = c.xe2[(size_t)bi * 64 + ch];
    fe2c[nt] = c.fe2[(size_t)bi * 64 + ch];
    yeac[nt] = c.yeav[b * 64 + ch];
    yemc[nt] = c.yemv[b * 64 + ch];
  }

  St st[5][2];
  for (int g = 0; g < 5; ++g) { st_init(st[g][0]); st_init(st[g][1]); }
  float velacc = 0.f, velpacc = 0.f;

  // prefetch first E tile rows for this wave (global_prefetch_b8 path)
  {
    const int j0f = wave * 16;
    __builtin_prefetch(c.E + ((size_t)bi * NN + j0f + (lane & 15)) * 64, 0, 1);
  }

  __syncthreads();

  for (int pass = 0; pass < (NN / 16) / EM_WAVES; ++pass) {
    const int jt = pass * EM_WAVES + wave;
    const int j0 = jt * 16;

    // prefetch next pass's E tile while this one is being produced
    if (pass + 1 < (NN / 16) / EM_WAVES) {
      const int j0n = j0 + 16 * EM_WAVES;
      __builtin_prefetch(c.E + ((size_t)bi * NN + j0n + (lane & 15)) * 64, 0, 1);
    }

    // per-j pair scalars (dist, dots, masks, positions)
    if (lane < 16) {
      int j = j0 + lane, bj = b * NN + j;
      float mj = c.nmask[bj], pmj = c.pmask[bj];
      float d2 = 0.f, dt = 0.f, d2p = 0.f, dtp = 0.f;
#pragma unroll
      for (int q3 = 0; q3 < 3; ++q3) {
        float pj = c.posm[bj * 3 + q3], nj = c.npos[bj * 3 + q3];
        float qj = c.pmm[bj * 3 + q3], nq = c.npp[bj * 3 + q3];
        float dd = posi[q3] - pj;  d2 += dd * dd;   dt += npi[q3] * nj;
        float ddp = ppi[q3] - qj;  d2p += ddp * ddp; dtp += nppi[q3] * nq;
        sJ[wave][lane][6 + q3] = pj;
        sJ[wave][lane][9 + q3] = qj;
      }
      sJ[wave][lane][0] = sqrtf(fmaxf(d2, 1e-12f));
      sJ[wave][lane][1] = dt;
      sJ[wave][lane][2] = sqrtf(fmaxf(d2p, 1e-12f));
      sJ[wave][lane][3] = dtp;
      sJ[wave][lane][4] = mj;
      sJ[wave][lane][5] = pmj;
    }
    __syncthreads();

    // dist1 / dist1_p tiles (relu(lin2 + node terms) * mask)
    for (int e = lane; e < 1024; e += 32) {
      int jj = e >> 6, ch = e & 63;
      int bj = b * NN + j0 + jj;
      float mj = sJ[wave][jj][4], pmj = sJ[wave][jj][5];
      float npj = c.normpos[bj], npjp = c.normp[bj];
      float v = c.ld1_w[ch] * sJ[wave][jj][0] + c.ld1_w[64 + ch] * sJ[wave][jj][1] +
                c.ld1_b[ch] + c.lnp1_w[ch] * normpos_i + c.lnp1_b[ch] +
                c.lnp2_w[ch] * npj + c.lnp2_b[ch];
      sD1[wave][jj * 64 + ch] = fmaxf(v, 0.f) * (m_i * mj);
      float vp = c.ld1p_w[ch] * sJ[wave][jj][2] + c.ld1p_w[64 + ch] * sJ[wave][jj][3] +
                 c.ld1p_b[ch] + c.lnp_w[ch] * normp_i + c.lnp_b[ch] +
                 c.lnp2p_w[ch] * npjp + c.lnp2p_b[ch];
      sD1P[wave][jj * 64 + ch] = fmaxf(vp, 0.f) * (pm_i * pmj);
    }
    __syncthreads();

    v8f acc[4];
    float Yreg[32], u[32], yp[32];

    // Y0 = in_E(E)
    init_acc(acc, c.inE_b, nl);
    gemm_tile(c.E + ((size_t)bi * NN + j0) * 64, c.inE_w, acc, lane);
#pragma unroll
    for (int nt = 0; nt < 4; ++nt) {
      int ch = nt * 16 + nl;
#pragma unroll
      for (int r = 0; r < 8; ++r) {
        int m = r + 8 * khh;
        float mj = sJ[wave][m][4], pmj = sJ[wave][m][5];
        float y0 = acc[nt][r];
        float xe1j = c.xe1[(size_t)(b * NN + j0 + m) * 64 + ch];
        float fe1j = c.fe1[(size_t)(b * NN + j0 + m) * 64 + ch];
        Yreg[nt * 8 + r] = y0 * xe1j * xe2c[nt] * (m_i * mj) +
                           y0 * fe1j * fe2c[nt] * (pm_i * pmj);
      }
    }
    // Y_pos = (Y*(1+dist_mul(d1)) + dist_add(d1)) * ee
    init_acc(acc, c.dme_b, nl);
    gemm_tile(&sD1[wave][0], c.dme_w, acc, lane);
#pragma unroll
    for (int nt = 0; nt < 4; ++nt)
#pragma unroll
      for (int r = 0; r < 8; ++r) u[nt * 8 + r] = Yreg[nt * 8 + r] * (1.f + acc[nt][r]);
    init_acc(acc, c.dae_b, nl);
    gemm_tile(&sD1[wave][0], c.dae_w, acc, lane);
#pragma unroll
    for (int nt = 0; nt < 4; ++nt)
#pragma unroll
      for (int r = 0; r < 8; ++r) {
        int m = r + 8 * khh;
        yp[nt * 8 + r] = (u[nt * 8 + r] + acc[nt][r]) * (m_i * sJ[wave][m][4]);
      }
    // Y_p and combine
    init_acc(acc, c.dmep_b, nl);
    gemm_tile(&sD1P[wave][0], c.dmep_w, acc, lane);
#pragma unroll
    for (int nt = 0; nt < 4; ++nt)
#pragma unroll
      for (int r = 0; r < 8; ++r) u[nt * 8 + r] = Yreg[nt * 8 + r] * (1.f + acc[nt][r]);
    init_acc(acc, c.daep_b, nl);
    gemm_tile(&sD1P[wave][0], c.daep_w, acc, lane);
#pragma unroll
    for (int nt = 0; nt < 4; ++nt)
#pragma unroll
      for (int r = 0; r < 8; ++r) {
        int m = r + 8 * khh;
        Yreg[nt * 8 + r] = yp[nt * 8 + r] +
                           (u[nt * 8 + r] + acc[nt][r]) * (pm_i * sJ[wave][m][5]);
      }
    // write Y_comb and Enew tiles
#pragma unroll
    for (int nt = 0; nt < 4; ++nt) {
      int ch = nt * 16 + nl;
#pragma unroll
      for (int r = 0; r < 8; ++r) {
        int m = r + 8 * khh;
        float yv = Yreg[nt * 8 + r];
        sYC[wave][m * 64 + ch] = yv;
        sEN[wave][m * 64 + ch] =
            (yv + yeac[nt] + yv * yemc[nt]) * (m_i * sJ[wave][m][4]);
      }
    }
    __syncthreads();

    // Eout = e_out(Enew) * ee
    init_acc(acc, c.eout_b, nl);
    gemm_tile(&sEN[wave][0], c.eout_w, acc, lane);
#pragma unroll
    for (int nt = 0; nt < 4; ++nt) {
      int ch = nt * 16 + nl;
#pragma unroll
      for (int r = 0; r < 8; ++r) {
        int m = r + 8 * khh;
        c.oE[((size_t)bi * NN + j0 + m) * 64 + ch] =
            acc[nt][r] * (m_i * sJ[wave][m][4]);
      }
    }

    // vel messages: e_pos2(relu(e_pos1(Y)))
    init_acc(acc, nullptr, nl);
    gemm_tile(&sYC[wave][0], c.ep1_w, acc, lane);
#pragma unroll
    for (int r = 0; r < 8; ++r) {
      float part = 0.f;
#pragma unroll
      for (int nt = 0; nt < 4; ++nt)
        part += fmaxf(acc[nt][r], 0.f) * c.ep2_w[nt * 16 + nl];
      part += __shfl_xor(part, 1, 32);
      part += __shfl_xor(part, 2, 32);
      part += __shfl_xor(part, 4, 32);
      part += __shfl_xor(part, 8, 32);
      if (nl < 3) {
        int m = r + 8 * khh;
        velacc += part * (posi[nl] - sJ[wave][m][6 + nl]);
      }
    }
    // vel_p messages: e_p2(relu(e_p1(Y)))
    init_acc(acc, nullptr, nl);
    gemm_tile(&sYC[wave][0], c.epp1_w, acc, lane);
#pragma unroll
    for (int r = 0; r < 8; ++r) {
      float part = 0.f;
#pragma unroll
      for (int nt = 0; nt < 4; ++nt)
        part += fmaxf(acc[nt][r], 0.f) * c.epp2_w[nt * 16 + nl];
      part += __shfl_xor(part, 1, 32);
      part += __shfl_xor(part, 2, 32);
      part += __shfl_xor(part, 4, 32);
      part += __shfl_xor(part, 8, 32);
      if (nl < 3) {
        int m = r + 8 * khh;
        velpacc += part * (ppi[nl] - sJ[wave][m][9 + nl]);
      }
    }

    // attention logits
    for (int it = 0; it < 4; ++it) {
      int task = lane + 32 * it;
      int jj = task >> 3, h = task & 7;
      int bj = b * NN + j0 + jj;
      float base = 0.f, afb = 0.f;
      for (int d = 0; d < 256; ++d) {
        float km = c.Km[(size_t)bj * 256 + d];
        base += km * sQmw[d * 8 + h];
        afb += km * sQfw[d * 8 + h];
      }
      float eatt = c.eatt_b[h], posatt = c.posatt_b[h], patt = c.patt_b[h];
      for (int ch = 0; ch < 64; ++ch) {
        eatt   += sEN[wave][jj * 64 + ch] * c.eatt_w[ch * 8 + h];
        posatt += sD1[wave][jj * 64 + ch] * c.posatt_w[ch * 8 + h];
        patt   += sD1P[wave][jj * 64 + ch] * c.patt_w[ch * 8 + h];
      }
      float mj = sJ[wave][jj][4], pmj = sJ[wave][jj][5];
      float eei = m_i * mj, pepi = pm_i * pmj;
      float av = base * 0.125f * eei;
      av *= (1.f + eatt);
      av *= (1.f + posatt);
      float apart = av * eei;
      av *= (1.f + patt);
      c.abuf[((size_t)bi * NN + j0 + jj) * 8 + h] =
          apart + av * pepi + afb * 0.125f * pepi;
    }

    // fused statistics (etox for Enew,dist1,dist1_p; etoy for Y,dist1)
    for (int jj = 0; jj < 16; ++jj) {
      float mj = sJ[wave][jj][4], pmj = sJ[wave][jj][5], eei = m_i * mj;
#pragma unroll
      for (int kk = 0; kk < 2; ++kk) {
        int ch = lane + kk * 32;
        float en = sEN[wave][jj * 64 + ch];
        float d1 = sD1[wave][jj * 64 + ch];
        float d1p = sD1P[wave][jj * 64 + ch];
        float yv = sYC[wave][jj * 64 + ch];
        st_upd(st[0][kk], en, mj);
        st_upd(st[1][kk], d1, mj);
        st_upd(st[2][kk], d1p, pmj);
        st_upd(st[3][kk], yv, eei);
        st_upd(st[4][kk], d1, eei);
      }
    }
    __syncthreads();
  }

  if (nl < 3) {
    atomicAdd(&sVel[nl], velacc);
    atomicAdd(&sVelp[nl], velpacc);
  }
#pragma unroll
  for (int g = 0; g < 5; ++g)
#pragma unroll
    for (int kk = 0; kk < 2; ++kk) {
      int ch = lane + kk * 32;
      sStats[wave][g][0][ch] = st[g][kk].s;
      sStats[wave][g][1][ch] = st[g][kk].sm;
      sStats[wave][g][2][ch] = st[g][kk].sq;
      sStats[wave][g][3][ch] = st[g][kk].mn;
      sStats[wave][g][4][ch] = st[g][kk].mx;
    }
  __syncthreads();
  if (tid < 64) {
    int ch = tid;
    for (int g = 0; g < 5; ++g) {
      float s = 0.f, sm = 0.f, sq = 0.f, mn = 1e30f, mx = -1e30f;
      for (int w2 = 0; w2 < EM_WAVES; ++w2) {
        s += sStats[w2][g][0][ch];
        sm += sStats[w2][g][1][ch];
        sq += sStats[w2][g][2][ch];
        mn = fminf(mn, sStats[w2][g][3][ch]);
        mx = fmaxf(mx, sStats[w2][g][4][ch]);
      }
      if (g < 3) {
        size_t o = (((size_t)bi * 3 + g) * 5) * 64;
        c.exst[o + ch] = s; c.exst[o + 64 + ch] = sm; c.exst[o + 128 + ch] = sq;
        c.exst[o + 192 + ch] = mn; c.exst[o + 256 + ch] = mx;
      } else {
        size_t o = (((size_t)bi * 2 + (g - 3)) * 5) * 64;
        c.eyst[o + ch] = s; c.eyst[o + 64 + ch] = sm; c.eyst[o + 128 + ch] = sq;
        c.eyst[o + 192 + ch] = mn; c.eyst[o + 256 + ch] = mx;
      }
    }
  }
  if (tid < 3) {
    c.vel[bi * 3 + tid] = sVel[tid] * m_i;
    c.velp[bi * 3 + tid] = sVelp[tid] * pm_i;
  }
}

// ================= softmax over j + weighted V =================
__global__ __launch_bounds__(256) void softmax_wv_kernel(Ctx c) {
  __shared__ float sA[2048];
  int bi = blockIdx.x, b = bi >> 8, t = threadIdx.x;
  for (int e = t; e < 2048; e += 256) {
    int j = e >> 3;
    float mj = c.nmask[b * NN + j];
    float av = c.abuf[(size_t)bi * 2048 + e];
    sA[e] = (mj > 0.f) ? av : -1e30f;
  }
  __syncthreads();
  if (t < 8) {
    float mx = -1e30f;
    for (int j = 0; j < 256; ++j) mx = fmaxf(mx, sA[j * 8 + t]);
    float s = 0.f;
    for (int j = 0; j < 256; ++j) {
      float e2 = expf(sA[j * 8 + t] - mx);
      sA[j * 8 + t] = e2; s += e2;
    }
    float inv = 1.f / s;
    for (int j = 0; j < 256; ++j) sA[j * 8 + t] *= inv;
  }
  __syncthreads();
  for (int h = 0; h < 8; ++h) {
    float accv = 0.f;
    for (int j = 0; j < 256; ++j)
      accv += sA[j * 8 + h] * c.Vv[(size_t)(b * NN + j) * 256 + t];
    c.wv[(size_t)bi * 2048 + h * 256 + t] = accv;
  }
}

// ================= out projection: (1024x2048)@(2048x256) via WMMA ========
// 16-node M tile per workgroup; 4 waves, each covers 64 output channels.
__global__ __launch_bounds__(128) void out_gemm_kernel(Ctx c) {
  const int tile = blockIdx.x;          // 64 tiles of 16 rows (b,i)
  const int wave = threadIdx.x >> 5;
  const int lane = threadIdx.x & 31;
  const int row0 = tile * 16;
  const int n0 = wave * 64;
  const int mA = lane & 15;
  const int kh2 = (lane >> 4) << 1;
  const int nl = lane & 15;
  const int khh = lane >> 4;
  const float* __restrict__ in = c.wv + (size_t)row0 * 2048;
  v8f acc[4];
#pragma unroll
  for (int nt = 0; nt < 4; ++nt) {
    float bb = c.out_b[n0 + nt * 16 + nl];
    v8f tv;
#pragma unroll
    for (int r = 0; r < 8; ++r) tv[r] = bb;
    acc[nt] = tv;
  }
  for (int kb = 0; kb < 2048; kb += 4) {
    v2f a;
    a.x = in[(size_t)mA * 2048 + kb + kh2];
    a.y = in[(size_t)mA * 2048 + kb + kh2 + 1];
#pragma unroll
    for (int nt = 0; nt < 4; ++nt) {
      v2f bf;
      bf.x = c.out_w[(size_t)(kb + kh2) * 256 + n0 + nt * 16 + nl];
      bf.y = c.out_w[(size_t)(kb + kh2 + 1) * 256 + n0 + nt * 16 + nl];
      acc[nt] = wmma4(a, bf, acc[nt]);
    }
  }
#pragma unroll
  for (int nt = 0; nt < 4; ++nt) {
    int ch = n0 + nt * 16 + nl;
#pragma unroll
    for (int r = 0; r < 8; ++r) {
      int m = r + 8 * khh;
      c.wvo[(size_t)(row0 + m) * 256 + ch] = acc[nt][r];
    }
  }
}

// ================= X update =================
__global__ __launch_bounds__(256) void x_update_kernel(Ctx c) {
  __shared__ float sCat[3][256];
  __shared__ float sNX[256];
  int bi = blockIdx.x, b = bi >> 8, t = threadIdx.x;
  float denj = 0.f, denp = 0.f;
  for (int j = 0; j < 256; ++j) {
    denj += c.nmask[b * NN + j];
    denp += c.pmask[b * NN + j];
  }
  if (t < 192) {
    int g = t >> 6, ch = t & 63;
    size_t o = (((size_t)bi * 3 + g) * 5) * 64;
    float s = c.exst[o + ch], sm = c.exst[o + 64 + ch], sq = c.exst[o + 128 + ch];
    float mn = c.exst[o + 192 + ch], mx = c.exst[o + 256 + ch];
    float den = (g == 2) ? denp : denj;
    float mean = s / den;
    float sd = (sq - 2.f * mean * sm + mean * mean * den) / den;
    sCat[g][ch] = mean; sCat[g][64 + ch] = mn; sCat[g][128 + ch] = mx; sCat[g][192 + ch] = sd;
  }
  __syncthreads();
  float exm = c.exm_b[t], pxm = c.pxm_b[t], ppxm = c.ppxm_b[t];
  for (int d = 0; d < 256; ++d) {
    exm  += sCat[0][d] * c.exm_w[d * 256 + t];
    pxm  += sCat[1][d] * c.pxm_w[d * 256 + t];
    ppxm += sCat[2][d] * c.ppxm_w[d * 256 + t];
  }
  float m_i = c.nmask[bi];
  float acc = c.wvo[(size_t)bi * 256 + t] * m_i;   // out(wV) * x_mask
  acc *= (1.f + exm);
  acc *= (1.f + pxm);
  acc = acc * (1.f + c.fxmv[(size_t)bi * 256 + t]) + c.fxav[(size_t)bi * 256 + t];
  acc *= (1.f + ppxm);
  float nx = acc * (1.f + c.yxmv[b * 256 + t]) + c.yxav[b * 256 + t];
  c.newx[(size_t)bi * 256 + t] = nx;
  sNX[t] = nx;
  __syncthreads();
  float xo = c.xout_b[t];
  for (int d = 0; d < 256; ++d) xo += sNX[d] * c.xout_w[d * 256 + t];
  c.oX[(size_t)bi * 256 + t] = xo * m_i;
}

// ================= y update =================
__global__ __launch_bounds__(256) void y_update_kernel(Ctx c) {
  __shared__ float sXc[1024], sEc[256], sDc[256], sNy[128], sH[128], sy[128];
  int b = blockIdx.x, t = threadIdx.x;
  if (t < 128) sy[t] = c.y[b * 128 + t];
  float deni = 0.f;
  for (int i2 = 0; i2 < 256; ++i2) deni += c.nmask[b * NN + i2];
  {  // xtoy over nodes, channel t
    float s = 0.f, sm = 0.f, sq = 0.f, mn = 1e30f, mx = -1e30f;
    for (int i2 = 0; i2 < 256; ++i2) {
      float x = c.newx[((size_t)(b * NN + i2)) * 256 + t];
      float mi = c.nmask[b * NN + i2];
      s += x; sm += x * mi; sq += x * x * mi;
      mn = fminf(mn, x + 1e6f * (1.f - mi));
      mx = fmaxf(mx, x - 1e6f * (1.f - mi));
    }
    float mean = s / deni;
    float sd = (sq - 2.f * mean * sm + mean * mean * deni) / deni;
    sXc[t] = mean; sXc[256 + t] = mn; sXc[512 + t] = mx; sXc[768 + t] = sd;
  }
  if (t < 128) {  // etoy combine over i
    int g = t >> 6, ch = t & 63;
    float s = 0.f, sm = 0.f, sq = 0.f, mn = 1e30f, mx = -1e30f;
    for (int i2 = 0; i2 < 256; ++i2) {
      size_t o = (((size_t)(b * NN + i2) * 2 + g) * 5) * 64;
      s += c.eyst[o + ch]; sm += c.eyst[o + 64 + ch]; sq += c.eyst[o + 128 + ch];
      mn = fminf(mn, c.eyst[o + 192 + ch]);
      mx = fmaxf(mx, c.eyst[o + 256 + ch]);
    }
    float den = deni * deni;
    float mean = s / den;
    float sd = (sq - 2.f * mean * sm + mean * mean * den) / den;
    float* dst = (g == 0) ? sEc : sDc;
    dst[ch] = mean; dst[64 + ch] = mn; dst[128 + ch] = mx; dst[192 + ch] = sd;
  }
  __syncthreads();
  if (t < 128) {
    float ny = c.yy_b[t] + c.xy_b[t] + c.ey_b[t] + c.dy_b[t];
    for (int d = 0; d < 128; ++d) ny += sy[d] * c.yy_w[d * 128 + t];
    for (int d = 0; d < 1024; ++d) ny += sXc[d] * c.xy_w[d * 128 + t];
    for (int d = 0; d < 256; ++d) ny += sEc[d] * c.ey_w[d * 128 + t];
    for (int d = 0; d < 256; ++d) ny += sDc[d] * c.dy_w[d * 128 + t];
    sNy[t] = ny;
  }
  __syncthreads();
  if (t < 128) {
    float h = c.y1_b[t];
    for (int d = 0; d < 128; ++d) h += sNy[d] * c.y1_w[d * 128 + t];
    sH[t] = fmaxf(h, 0.f);
  }
  __syncthreads();
  if (t < 128) {
    float o = c.y2_b[t];
    for (int d = 0; d < 128; ++d) o += sH[d] * c.y2_w[d * 128 + t];
    c.oy[b * 128 + t] = o;
  }
}

// ================= pos attention pre (Qp/Kp/Vp from vel) =================
__global__ __launch_bounds__(64) void posqkv_kernel(Ctx c) {
  int bi = blockIdx.x, t = threadIdx.x;
  float m = c.nmask[bi];
  float v0 = c.vel[bi * 3 + 0], v1 = c.vel[bi * 3 + 1], v2 = c.vel[bi * 3 + 2];
  c.Qp[(size_t)bi * 64 + t] =
      (v0 * c.qpos_w[t] + v1 * c.qpos_w[64 + t] + v2 * c.qpos_w[128 + t] + c.qpos_b[t]) * m;
  c.Kp[(size_t)bi * 64 + t] =
      (v0 * c.kpos_w[t] + v1 * c.kpos_w[64 + t] + v2 * c.kpos_w[128 + t] + c.kpos_b[t]) * m;
  c.Vp[(size_t)bi * 64 + t] =
      (v0 * c.vpos_w[t] + v1 * c.vpos_w[64 + t] + v2 * c.vpos_w[128 + t] + c.vpos_b[t]) * m;
}

// ================= pos attention =================
__global__ __launch_bounds__(256) void pos_attn_kernel(Ctx c) {
  __shared__ float sQpw[512];
  __shared__ float sAp[2048];
  __shared__ float sWvp[512];
  int bi = blockIdx.x, b = bi >> 8, t = threadIdx.x;
  float m_i = c.nmask[bi], pm_i = c.pmask[bi];
  for (int e = t; e < 512; e += 256) {
    int d = e >> 3, h = e & 7;
    sQpw[e] = c.Qp[(size_t)bi * 64 + d] * c.apos_w[d * 8 + h];
  }
  __syncthreads();
  {
    int j = t, bj = b * NN + j;
    float mj = c.nmask[bj], pmj = c.pmask[bj];
    float d2 = 0.f;
    for (int q3 = 0; q3 < 3; ++q3) {
      float dd = c.pmm[bi * 3 + q3] - c.pmm[bj * 3 + q3];
      d2 += dd * dd;
    }
    float dpn = sqrtf(fmaxf(d2, 1e-12f)) * (pm_i * pmj);
    for (int h = 0; h < 8; ++h) {
      float apv = 0.f;
      for (int d = 0; d < 64; ++d) apv += c.Kp[(size_t)bj * 64 + d] * sQpw[d * 8 + h];
      apv = apv * 0.125f * (m_i * mj) -
            (c.dpos_w[h] * dpn + c.dpos_b[h]) * (pm_i * pmj);
      sAp[j * 8 + h] = (mj > 0.f) ? apv : -1e30f;
    }
  }
  __syncthreads();
  if (t < 8) {
    float mx = -1e30f;
    for (int j = 0; j < 256; ++j) mx = fmaxf(mx, sAp[j * 8 + t]);
    float s = 0.f;
    for (int j = 0; j < 256; ++j) {
      float e2 = expf(sAp[j * 8 + t] - mx);
      sAp[j * 8 + t] = e2; s += e2;
    }
    float inv = 1.f / s;
    for (int j = 0; j < 256; ++j) sAp[j * 8 + t] *= inv;
  }
  __syncthreads();
  for (int o = t; o < 512; o += 256) {
    int h = o >> 6, d = o & 63;
    float acc = 0.f;
    for (int j = 0; j < 256; ++j)
      acc += sAp[j * 8 + h] * c.Vp[(size_t)(b * NN + j) * 64 + d];
    sWvp[o] = acc;
  }
  __syncthreads();
  if (t < 3) {
    float o2 = c.outpos_b[t];
    for (int d = 0; d < 512; ++d) o2 += sWvp[d] * c.outpos_w[d * 3 + t];
    c.wvpo[bi * 3 + t] = o2 * m_i;
  }
}

// ================= mean removal for vel outputs =================
__global__ __launch_bounds__(256) void final_vel_kernel(Ctx c) {
  __shared__ float sS[3], sSP[3];
  int b = blockIdx.x, t = threadIdx.x;
  float denN = 0.f, denP = 0.f;
  for (int i2 = 0; i2 < 256; ++i2) {
    denN += c.nmask[b * NN + i2];
    denP += c.pmask[b * NN + i2];
  }
  if (t < 3) {
    float s = 0.f, sp = 0.f;
    for (int i2 = 0; i2 < 256; ++i2) {
      s += c.wvpo[(size_t)(b * NN + i2) * 3 + t];
      sp += c.velp[(size_t)(b * NN + i2) * 3 + t];
    }
    sS[t] = s / denN;
    sSP[t] = sp / denP;
  }
  __syncthreads();
  for (int e = t; e < 768; e += 256) {
    int i2 = e / 3, q3 = e % 3;
    float mi = c.nmask[b * NN + i2], pmi = c.pmask[b * NN + i2];
    c.ovel[(size_t)(b * NN + i2) * 3 + q3] =
        c.wvpo[(size_t)(b * NN + i2) * 3 + q3] - sS[q3] * mi;
    c.ovelp[(size_t)(b * NN + i2) * 3 + q3] =
        c.velp[(size_t)(b * NN + i2) * 3 + q3] - sSP[q3] * pmi;
  }
}

// ================= host launcher =================
extern "C" void kernel_launch(void* const* d_in, const int* in_sizes, int n_in,
                              void* d_out, int out_size, void* d_ws, size_t ws_size,
                              hipStream_t stream) {
  (void)in_sizes; (void)n_in; (void)out_size; (void)ws_size;
  Ctx c;
  c.X = (const float*)d_in[0];  c.E = (const float*)d_in[1];
  c.y = (const float*)d_in[2];  c.pos = (const float*)d_in[3];
  c.f = (const float*)d_in[4];  c.p = (const float*)d_in[5];
  c.nmask = (const float*)d_in[6]; c.pmask = (const float*)d_in[7];

  int ip = 8;
  auto nxt = [&]() { return (const float*)d_in[ip++]; };
  c.inE_w = nxt();  c.inE_b = nxt();
  c.xem1_w = nxt(); c.xem1_b = nxt();
  c.xem2_w = nxt(); c.xem2_b = nxt();
  c.fem1_w = nxt(); c.fem1_b = nxt();
  c.fem2_w = nxt(); c.fem2_b = nxt();
  c.ld1_w = nxt();  c.ld1_b = nxt();
  c.lnp1_w = nxt(); c.lnp1_b = nxt();
  c.lnp2_w = nxt(); c.lnp2_b = nxt();
  c.dae_w = nxt();  c.dae_b = nxt();
  c.dme_w = nxt();  c.dme_b = nxt();
  c.ld1p_w = nxt(); c.ld1p_b = nxt();
  c.lnp_w = nxt();  c.lnp_b = nxt();
  c.lnp2p_w = nxt(); c.lnp2p_b = nxt();
  c.daep_w = nxt(); c.daep_b = nxt();
  c.dmep_w = nxt(); c.dmep_b = nxt();
  c.k_w = nxt();    c.k_b = nxt();
  c.q_w = nxt();    c.q_b = nxt();
  c.v_w = nxt();    c.v_b = nxt();
  c.a_w = nxt();
  c.out_w = nxt();  c.out_b = nxt();
  c.qf_w = nxt();   c.qf_b = nxt();
  c.af_w = nxt();
  c.kpos_w = nxt(); c.kpos_b = nxt();
  c.qpos_w = nxt(); c.qpos_b = nxt();
  c.vpos_w = nxt(); c.vpos_b = nxt();
  c.apos_w = nxt();
  c.dpos_w = nxt(); c.dpos_b = nxt();
  c.outpos_w = nxt(); c.outpos_b = nxt();
  c.eatt_w = nxt(); c.eatt_b = nxt();
  c.posatt_w = nxt(); c.posatt_b = nxt();
  c.patt_w = nxt(); c.patt_b = nxt();
  c.exm_w = nxt();  c.exm_b = nxt();
  c.pxm_w = nxt();  c.pxm_b = nxt();
  c.ppxm_w = nxt(); c.ppxm_b = nxt();
  c.yem_w = nxt();  c.yem_b = nxt();
  c.yea_w = nxt();  c.yea_b = nxt();
  c.yxm_w = nxt();  c.yxm_b = nxt();
  c.yxa_w = nxt();  c.yxa_b = nxt();
  c.fxm_w = nxt();  c.fxm_b = nxt();
  c.fxa_w = nxt();  c.fxa_b = nxt();
  c.yy_w = nxt();   c.yy_b = nxt();
  c.xy_w = nxt();   c.xy_b = nxt();
  c.ey_w = nxt();   c.ey_b = nxt();
  c.dy_w = nxt();   c.dy_b = nxt();
  c.ep1_w = nxt();  c.ep2_w = nxt();
  c.epp1_w = nxt(); c.epp2_w = nxt();
  c.xout_w = nxt(); c.xout_b = nxt();
  c.eout_w = nxt(); c.eout_b = nxt();
  c.y1_w = nxt();   c.y1_b = nxt();
  c.y2_w = nxt();   c.y2_b = nxt();

  float* w = (float*)d_ws;
  c.xe1 = w + OFF_XE1;   c.xe2 = w + OFF_XE2;
  c.fe1 = w + OFF_FE1;   c.fe2 = w + OFF_FE2;
  c.Qm = w + OFF_QM;     c.Km = w + OFF_KM;     c.Vv = w + OFF_VV;
  c.Qf = w + OFF_QF;     c.fxmv = w + OFF_FXM;  c.fxav = w + OFF_FXA;
  c.posm = w + OFF_POSM; c.pmm = w + OFF_PMM;
  c.npos = w + OFF_NPOS; c.npp = w + OFF_NPP;
  c.normpos = w + OFF_NORMPOS; c.normp = w + OFF_NORMP;
  c.yeav = w + OFF_YEA;  c.yemv = w + OFF_YEM;
  c.yxmv = w + OFF_YXM;  c.yxav = w + OFF_YXA;
  c.abuf = w + OFF_ABUF; c.exst = w + OFF_EXST; c.eyst = w + OFF_EYST;
  c.vel = w + OFF_VEL;   c.velp = w + OFF_VELP;
  c.wv = w + OFF_WV;     c.newx = w + OFF_NEWX;
  c.Qp = w + OFF_QP;     c.Kp = w + OFF_KP;     c.Vp = w + OFF_VP;
  c.wvpo = w + OFF_WVPO; c.wvo = w + OFF_WVO;

  float* o = (float*)d_out;
  c.oX = o + OOUT_X; c.oE = o + OOUT_E; c.oy = o + OOUT_Y;
  c.ovel = o + OOUT_VEL; c.ovelp = o + OOUT_VELP;

  // f passthrough output
  hipMemcpyAsync(o + OOUT_F, (const void*)c.f, (size_t)4 * 256 * 256 * sizeof(float),
                 hipMemcpyDeviceToDevice, stream);

  batch_pre_kernel<<<4, 256, 0, stream>>>(c);
  node_pre_kernel<<<1024, 256, 0, stream>>>(c);
  e_main_kernel<<<1024, EM_THREADS, 0, stream>>>(c);
  softmax_wv_kernel<<<1024, 256, 0, stream>>>(c);
  out_gemm_kernel<<<64, 128, 0, stream>>>(c);
  x_update_kernel<<<1024, 256, 0, stream>>>(c);
  y_update_kernel<<<4, 256, 0, stream>>>(c);
  posqkv_kernel<<<1024, 64, 0, stream>>>(c);
  pos_attn_kernel<<<1024, 256, 0, stream>>>(c);
  final_vel_kernel<<<4, 256, 0, stream>>>(c);
}